// TAdaAggregation_16209206575444
// MI455X (gfx1250) — compile-verified
//
#include <hip/hip_runtime.h>

typedef __attribute__((ext_vector_type(2))) float v2f;
typedef __attribute__((ext_vector_type(8))) float v8f;
typedef unsigned int v4u __attribute__((ext_vector_type(4)));
typedef int v4i __attribute__((ext_vector_type(4)));
typedef int v8i __attribute__((ext_vector_type(8)));

#define N_  64
#define C_  256
#define T_  64
#define V_  25
#define U_  256
#define XS  40   // LDS row stride in dwords: 32 data (TDM zero-fills v=25..31) + 8 TDM pad

__global__ __launch_bounds__(256, 2)
void tada_wmma_f32_kernel(const float* __restrict__ x,
                          const float* __restrict__ alpha,
                          const float* __restrict__ w,
                          float* __restrict__ out) {
    __shared__ float Xs[C_ * XS];   // 40 KB, filled by TDM: Xs[c*40 + v], v=25..31 zero
    __shared__ float As[U_];        // 1 KB

    const int blk = blockIdx.x;     // one (n,t) pair per block
    const int n   = blk / T_;
    const int t   = blk - n * T_;
    const int tid = threadIdx.x;

    const float* xp = x + ((size_t)n * C_ * T_ + t) * V_;   // x[n, c, t, v]: + c*1600 + v

    // ---- TDM: one DMA moves the whole 256x25 X tile (zero-padded to 32, LDS rows padded to 40) ----
    if (tid < 32) {   // wave 0 issues; EXEC is ignored by TDM, so gate by wave
        const unsigned ldsa = (unsigned)(uintptr_t)&Xs[0];
        const unsigned long long ga = (unsigned long long)(uintptr_t)xp;

        v4u g0;
        g0[0] = 1u;                                   // count=1, user mode, no gather
        g0[1] = ldsa;                                 // lds_addr
        g0[2] = (unsigned)ga;                         // global_addr[31:0]
        g0[3] = (unsigned)(ga >> 32) | 0x80000000u;   // global_addr[56:32] | type=2<<30

        v8i g1;
        g1[0] = (int)((2u << 16)      // data_size = 4B
                    | (1u << 20)      // pad_enable
                    | (4u << 22)      // pad_interval: 32 dwords (end of each 32-dword row)
                    | (7u << 25));    // pad_amount:   8 dwords  -> LDS row stride 40
        g1[1] = (int)(25u << 16);     // tensor_dim0 = 25 (reads at v>=25 return zero)
        g1[2] = (int)(256u << 16);    // tensor_dim1 = 256 (dim0 hi16 = 0)
        g1[3] = (int)(32u << 16);     // tile_dim0 = 32 (dim1 hi16 = 0)
        g1[4] = 256;                  // tile_dim1 = 256, tile_dim2 = 0
        g1[5] = 1600;                 // tensor_dim0_stride = T_*V_ = 1600 dwords
        g1[6] = 0;                    // stride0 hi16 = 0, dim1_stride lo16 = 0
        g1[7] = 0;

        v4i g2 = {0, 0, 0, 0};        // no dims 2..4
        v4i g3 = {0, 0, 0, 0};
        v8i g4 = {0, 0, 0, 0, 0, 0, 0, 0};   // unused trailing group on this toolchain

        __builtin_amdgcn_tensor_load_to_lds(g0, g1, g2, g3, g4, 0);
        __builtin_amdgcn_s_wait_tensorcnt(0);
    }

    // ---- stage alpha[n, u, t] ----
    As[tid] = alpha[((size_t)n * U_ + tid) * T_ + t];
    __syncthreads();

    const int wave  = tid >> 5;     // 0..7
    const int lane  = tid & 31;
    const int lhalf = lane >> 4;    // 0 or 1
    const int lmod  = lane & 15;

    // Per-lane LDS byte address of B fragment base at k0 = 0.
    const unsigned blane = (unsigned)(uintptr_t)&Xs[0] + ((2 * lhalf) * XS + lmod) * 4;

    // B fragment loads pinned as paired 2addr ds loads: each yields {k, k+1} in a VGPR pair.
#define LOAD_B(b00_, b01_, b10_, b11_, addr_)                                             \
    asm volatile("ds_load_2addr_b32 %0, %1 offset1:40"              : "=v"(b00_) : "v"(addr_)); \
    asm volatile("ds_load_2addr_b32 %0, %1 offset0:16 offset1:56"   : "=v"(b01_) : "v"(addr_)); \
    asm volatile("ds_load_2addr_b32 %0, %1 offset0:160 offset1:200" : "=v"(b10_) : "v"(addr_)); \
    asm volatile("ds_load_2addr_b32 %0, %1 offset0:176 offset1:216" : "=v"(b11_) : "v"(addr_))

    // Wait until the 4 oldest DS loads (the fragments consumed now) are done;
    // the 4 just-prefetched ones stay in flight. Tied to operands to pin ordering.
#define WAIT_B(b00_, b01_, b10_, b11_)                                                    \
    asm volatile("s_wait_dscnt 0x4" : "+v"(b00_), "+v"(b01_), "+v"(b10_), "+v"(b11_))

    // Each wave: u-tiles {wave, wave+8}, both v-tiles each.
    for (int ui = 0; ui < 2; ++ui) {
        const int u0 = (wave + ui * 8) * 16;

        v8f acc0a = {}, acc0b = {};
        v8f acc1a = {}, acc1b = {};

        const float* wrow = w + (size_t)(u0 + lmod) * C_ + 2 * lhalf;

        // prologue: fragments for k0 = 0
        v2f a0 = *(const v2f*)(wrow);
        v2f a1 = *(const v2f*)(wrow + 4);
        v2f b00, b01, b10, b11;
        LOAD_B(b00, b01, b10, b11, blane);

        #pragma unroll 2
        for (int k0 = 0; k0 < C_; k0 += 8) {
            const int kn = (k0 + 8) & (C_ - 1);          // wrap keeps prefetch in-bounds
            const unsigned bn = blane + (unsigned)kn * (XS * 4);
            v2f nb00, nb01, nb10, nb11;
            LOAD_B(nb00, nb01, nb10, nb11, bn);
            v2f na0 = *(const v2f*)(wrow + kn);
            v2f na1 = *(const v2f*)(wrow + kn + 4);

            WAIT_B(b00, b01, b10, b11);
            acc0a = __builtin_amdgcn_wmma_f32_16x16x4_f32(false, a0, false, b00,
                                                          (short)0, acc0a, false, false);
            acc1a = __builtin_amdgcn_wmma_f32_16x16x4_f32(false, a0, false, b01,
                                                          (short)0, acc1a, false, false);
            acc0b = __builtin_amdgcn_wmma_f32_16x16x4_f32(false, a1, false, b10,
                                                          (short)0, acc0b, false, false);
            acc1b = __builtin_amdgcn_wmma_f32_16x16x4_f32(false, a1, false, b11,
                                                          (short)0, acc1b, false, false);

            a0 = na0; a1 = na1;
            b00 = nb00; b01 = nb01; b10 = nb10; b11 = nb11;
        }

        // ---- scale by alpha and store (single divergent region for the ragged v-tile) ----
        float sv[8];
        #pragma unroll
        for (int r = 0; r < 8; ++r) sv[r] = As[u0 + r + 8 * lhalf];

        float* op = out + (((size_t)n * U_ + (u0 + 8 * lhalf)) * T_ + t) * V_ + lmod;
        #pragma unroll
        for (int r = 0; r < 8; ++r)
            op[(size_t)r * (T_ * V_)] = sv[r] * (acc0a[r] + acc0b[r]);
        if (lmod < V_ - 16) {
            #pragma unroll
            for (int r = 0; r < 8; ++r)
                op[(size_t)r * (T_ * V_) + 16] = sv[r] * (acc1a[r] + acc1b[r]);
        }
    }
#undef LOAD_B
#undef WAIT_B
}

extern "C" void kernel_launch(void* const* d_in, const int* in_sizes, int n_in,
                              void* d_out, int out_size, void* d_ws, size_t ws_size,
                              hipStream_t stream) {
    const float* x     = (const float*)d_in[0];
    const float* alpha = (const float*)d_in[1];
    const float* w     = (const float*)d_in[2];
    float* out = (float*)d_out;
    (void)in_sizes; (void)n_in; (void)out_size; (void)d_ws; (void)ws_size;

    dim3 grid(N_ * T_);   // 4096 blocks, one (n,t) each
    dim3 block(256);      // 8 waves
    tada_wmma_f32_kernel<<<grid, block, 0, stream>>>(x, alpha, w, out);
}